// MultiIndexSelect_48017734370006
// MI455X (gfx1250) — compile-verified
//
#include <hip/hip_runtime.h>

// MultiIndexSelect on MI455X (gfx1250): pure bandwidth kernel.
// Path: GLOBAL_LOAD_ASYNC_TO_LDS_B128 -> s_wait_asynccnt -> GLOBAL_STORE_ASYNC_FROM_LDS_B128.
// One wave32 per 1KB row; 2 async ops each direction (32 lanes x 16B each).
// Grid is 2D (rows x layers) so there is NO integer division in the kernel.

#define COLS 256  // C in the reference; row = 1024 bytes

__global__ __launch_bounds__(256) void mis_async_row_copy(
    const int* __restrict__ idx_from0, const int* __restrict__ idx_to0,
    const int* __restrict__ idx_from1, const int* __restrict__ idx_to1,
    const float* __restrict__ mat0, const float* __restrict__ mat1,
    float* __restrict__ out, int twoK, int K, int N)
{
    // 8 waves per block, 1 row (1KB) of LDS staging per wave.
    __shared__ float lds[8 * COLS];

    const unsigned lane = threadIdx.x & 31u;
    const unsigned wib  = threadIdx.x >> 5;            // wave index in block
    const int r0 = (int)(blockIdx.x * 8u + wib);       // output row within layer
    if (r0 >= twoK) return;                            // wave-uniform; EXEC stays all-1s
    const int layer = (int)blockIdx.y;

    // Uniform pointer-select (no exec-masked dual paths, single index loads).
    const bool second   = (r0 >= K);
    const int  r        = second ? (r0 - K) : r0;
    const int* idxF     = second ? idx_from1 : idx_from0;
    const int* idxT     = second ? idx_to1   : idx_to0;
    const float* srcMat = second ? mat1      : mat0;

    const int from = idxF[r];
    const int to   = idxT[r];

    const float* srcRow = srcMat + ((size_t)layer * (size_t)N    + (size_t)from) * COLS;
    float*       dstRow = out    + ((size_t)layer * (size_t)twoK + (size_t)to)   * COLS;

    // LDS byte offset for this lane's 16B slot (generic ptr low 32 bits == LDS offset).
    unsigned ldsAddr = (unsigned)(unsigned long long)(&lds[wib * COLS]) + lane * 16u;
    const float* g = srcRow + lane * 4u;  // 16 bytes per lane
    float*       d = dstRow + lane * 4u;

    // Gather row into LDS (async, ASYNCcnt). The 24-bit inst offset applies to
    // BOTH the global and LDS addresses, so the same registers cover [512,1024).
    asm volatile("global_load_async_to_lds_b128 %0, %1, off"
                 :: "v"(ldsAddr), "v"(g) : "memory");
    asm volatile("global_load_async_to_lds_b128 %0, %1, off offset:512"
                 :: "v"(ldsAddr), "v"(g) : "memory");

    // Wait for this wave's async loads before the stores read LDS.
    asm volatile("s_wait_asynccnt 0" ::: "memory");

    // Scatter row from LDS to out (async).
    asm volatile("global_store_async_from_lds_b128 %0, %1, off"
                 :: "v"(d), "v"(ldsAddr) : "memory");
    asm volatile("global_store_async_from_lds_b128 %0, %1, off offset:512"
                 :: "v"(d), "v"(ldsAddr) : "memory");

    // S_ENDPGM performs an implicit wait-idle, covering the outstanding async stores.
}

extern "C" void kernel_launch(void* const* d_in, const int* in_sizes, int n_in,
                              void* d_out, int out_size, void* d_ws, size_t ws_size,
                              hipStream_t stream) {
    (void)n_in; (void)d_ws; (void)ws_size;

    const int*   idx_from0 = (const int*)d_in[0];
    const int*   idx_to0   = (const int*)d_in[1];
    const int*   idx_from1 = (const int*)d_in[2];
    const int*   idx_to1   = (const int*)d_in[3];
    const float* mat0      = (const float*)d_in[4];
    const float* mat1      = (const float*)d_in[5];
    float*       out       = (float*)d_out;

    const int K    = in_sizes[0];                 // 30000
    const int twoK = 2 * K;                       // 60000 output rows per layer
    const int L    = out_size / (twoK * COLS);    // 4 layers
    const int N    = in_sizes[4] / (L * COLS);    // 60000 source rows per layer

    const int wavesPerBlock = 8;                  // 256 threads, wave32
    dim3 grid((unsigned)((twoK + wavesPerBlock - 1) / wavesPerBlock), (unsigned)L, 1);

    mis_async_row_copy<<<grid, 256, 0, stream>>>(
        idx_from0, idx_to0, idx_from1, idx_to1, mat0, mat1, out,
        twoK, K, N);
}